// ReadSetClassifier_86586540687904
// MI455X (gfx1250) — compile-verified
//
#include <hip/hip_runtime.h>
#include <hip/hip_bf16.h>

// ---------------- problem constants ----------------
#define NF_READ     11
#define NF_INFO     9
#define BATCH       16384
#define TOTAL_READS 524288
#define NUM_SEG     (2 * BATCH)   // 32768
#define HDIM        64
#define MAX_ALT     10

typedef float v2f __attribute__((ext_vector_type(2)));
typedef float v8f __attribute__((ext_vector_type(8)));

// fp32 WMMA: D(16x16) = A(16x4) * B(4x16) + C, full fp32 precision.
__device__ __forceinline__ v8f wmma4(v2f a, v2f b, v8f c) {
    return __builtin_amdgcn_wmma_f32_16x16x4_f32(false, a, false, b, (short)0, c,
                                                 false, false);
}

__device__ __forceinline__ float sigmoidf_(float x) {
    return 1.0f / (1.0f + __expf(-x));
}

// =====================================================================
// Encoder kernel: X(N x NF) -> relu(X W1 + b1) -> sigmoid(h W2 + b2)
// Layer 1 (NF->64, K tiny) in VALU; layer 2 (64->64) via fp32 WMMA.
// Block = 256 threads = 8 waves, 16 rows per wave -> 128 rows / block.
// SEGMENT=true: run-length-aggregated atomic scatter into per-segment
//               sums + counts (phi path; segment ids are sorted).
// SEGMENT=false: direct store of sigmoid output (omega path).
// =====================================================================
template <int NF, bool SEGMENT>
__global__ __launch_bounds__(256) void encoder_kernel(
    const float* __restrict__ X,
    const float* __restrict__ W1,   // NF x 64 (row-major, f-major)
    const float* __restrict__ B1,   // 64
    const float* __restrict__ W2,   // 64 x 64 (row-major, k-major)
    const float* __restrict__ B2,   // 64
    const int*  __restrict__ segid, // TOTAL_READS (SEGMENT only)
    float* __restrict__ out,        // SEGMENT: sums[NUM_SEG][64]; else [N][64]
    float* __restrict__ counts)     // SEGMENT only: NUM_SEG
{
    __shared__ __align__(16) float sW1[NF * HDIM];
    __shared__ __align__(16) float sB1[HDIM];
    __shared__ __align__(16) float sW2t[HDIM * HDIM]; // column-major: [n][k]
    __shared__ __align__(16) float sB2[HDIM];
    __shared__ __align__(16) float sX[128 * NF];      // staged input tile

    const int tid = threadIdx.x;
    // coalesced stage of the 128-row input tile
    {
        const float* Xt = X + (size_t)blockIdx.x * 128 * NF;
        for (int i = tid; i < 128 * NF; i += 256) sX[i] = Xt[i];
    }
    for (int i = tid; i < NF * HDIM; i += 256) sW1[i] = W1[i];
    for (int i = tid; i < HDIM * HDIM; i += 256) {
        int k = i / HDIM, n = i % HDIM;
        sW2t[n * HDIM + k] = W2[i];                   // transpose into LDS
    }
    if (tid < HDIM) { sB1[tid] = B1[tid]; sB2[tid] = B2[tid]; }
    __syncthreads();

    const int wave    = tid >> 5;
    const int lane    = tid & 31;
    const int half    = lane >> 4;          // 0: K 0/1, 1: K 2/3
    const int n_lane  = lane & 15;
    const int rowbase = blockIdx.x * 128 + wave * 16;
    const int mloc    = wave * 16 + n_lane; // tile-local row owned by this lane

    // ---- layer 1 (VALU): each lane produces its 32 hidden cols in A-format
    float x[NF];
#pragma unroll
    for (int f = 0; f < NF; ++f) x[f] = sX[mloc * NF + f];

    v2f afrag[16];
#pragma unroll
    for (int j = 0; j < 16; ++j) {
        const int c0 = 4 * j + 2 * half;
        float s0 = sB1[c0], s1 = sB1[c0 + 1];
#pragma unroll
        for (int f = 0; f < NF; ++f) {
            s0 = fmaf(x[f], sW1[f * HDIM + c0], s0);
            s1 = fmaf(x[f], sW1[f * HDIM + c0 + 1], s1);
        }
        afrag[j] = v2f{fmaxf(s0, 0.0f), fmaxf(s1, 0.0f)};
    }

    // ---- layer 2 (64->64) via fp32 WMMA: 16 k-steps x 4 n-tiles
    v8f acc[4];
#pragma unroll
    for (int nt = 0; nt < 4; ++nt)
        acc[nt] = v8f{0.f, 0.f, 0.f, 0.f, 0.f, 0.f, 0.f, 0.f};

#pragma unroll 4
    for (int kk = 0; kk < 16; ++kk) {
        const int k0 = 4 * kk + 2 * half;
#pragma unroll
        for (int nt = 0; nt < 4; ++nt) {
            const int n = nt * 16 + n_lane;
            v2f b = *(const v2f*)&sW2t[n * HDIM + k0];   // ds_load_b64
            acc[nt] = wmma4(afrag[kk], b, acc[nt]);
        }
    }

    // ---- epilogue: bias + sigmoid, then scatter / store.
    // C layout: this lane holds rows (rowbase + 8*half + r), r = 0..7,
    // which are CONSECUTIVE sorted rows -> run-length aggregate atomics.
    int segs[8];
#pragma unroll
    for (int r = 0; r < 8; ++r) {
        const int row = rowbase + 8 * half + r;
        if (SEGMENT) segs[r] = segid[row];
        else         segs[r] = row;
    }
#pragma unroll
    for (int nt = 0; nt < 4; ++nt) {
        const int n = nt * 16 + n_lane;
        const float bias = sB2[n];
        if (SEGMENT) {
            int   curseg = segs[0];
            float run    = 0.0f;
#pragma unroll
            for (int r = 0; r < 8; ++r) {
                const float v = sigmoidf_(acc[nt][r] + bias);
                if (segs[r] == curseg) {
                    run += v;
                } else {
                    atomicAdd(&out[curseg * HDIM + n], run);
                    curseg = segs[r];
                    run    = v;
                }
            }
            atomicAdd(&out[curseg * HDIM + n], run);
        } else {
#pragma unroll
            for (int r = 0; r < 8; ++r)
                out[segs[r] * HDIM + n] = sigmoidf_(acc[nt][r] + bias);
        }
    }
    if (SEGMENT && lane < 16)
        atomicAdd(&counts[segid[rowbase + lane]], 1.0f);
}

// =====================================================================
// rho kernel: concat(ref_mean, alt_mean, omega) (B x 192)
//   -> relu(. Wr1 + br1) (128) -> . Wr2 + br2 (64), all via fp32 WMMA.
// Block = 128 threads = 4 waves, 16 rows/wave -> 64 rows/block.
// =====================================================================
__global__ __launch_bounds__(128) void rho_kernel(
    const float* __restrict__ sums,    // NUM_SEG x 64
    const float* __restrict__ counts,  // NUM_SEG
    const float* __restrict__ omega,   // B x 64
    const float* __restrict__ Wr1,     // 192 x 128
    const float* __restrict__ br1,     // 128
    const float* __restrict__ Wr2,     // 128 x 64
    const float* __restrict__ br2,     // 64
    float* __restrict__ agg)           // B x 64
{
    __shared__ __align__(16) float sC[64 * 192];   // concat tile, reused for h1

    const int tid     = threadIdx.x;
    const int rowbase = blockIdx.x * 64;

    // stage concat = [ref_mean | alt_mean | omega] into LDS
    for (int idx = tid; idx < 64 * 192; idx += 128) {
        const int rl = idx / 192, c = idx % 192;
        const int r  = rowbase + rl;
        float v;
        if (c < 64)
            v = sums[r * HDIM + c] / fmaxf(counts[r], 1.0f);
        else if (c < 128)
            v = sums[(BATCH + r) * HDIM + (c - 64)] / fmaxf(counts[BATCH + r], 1.0f);
        else
            v = omega[r * HDIM + (c - 128)];
        sC[rl * 192 + c] = v;
    }
    __syncthreads();

    const int wave   = tid >> 5;
    const int lane   = tid & 31;
    const int half   = lane >> 4;
    const int n_lane = lane & 15;
    const int mloc   = wave * 16 + n_lane;

    // ---- layer rho1: K=192 (48 steps), N=128 (8 tiles)
    v8f acc1[8];
#pragma unroll
    for (int nt = 0; nt < 8; ++nt)
        acc1[nt] = v8f{0.f, 0.f, 0.f, 0.f, 0.f, 0.f, 0.f, 0.f};

#pragma unroll 2
    for (int kk = 0; kk < 48; ++kk) {
        const int k0 = 4 * kk + 2 * half;
        v2f a = *(const v2f*)&sC[mloc * 192 + k0];
#pragma unroll
        for (int nt = 0; nt < 8; ++nt) {
            const int n = nt * 16 + n_lane;
            v2f b = v2f{Wr1[k0 * 128 + n], Wr1[(k0 + 1) * 128 + n]};
            acc1[nt] = wmma4(a, b, acc1[nt]);
        }
    }
    __syncthreads();  // all concat reads done before overwrite

    // bias + relu, repack h1 (64 rows x 128) into LDS
#pragma unroll
    for (int nt = 0; nt < 8; ++nt) {
        const int n = nt * 16 + n_lane;
        const float bias = br1[n];
#pragma unroll
        for (int i = 0; i < 8; ++i) {
            const int rl = wave * 16 + 8 * half + i;
            sC[rl * 128 + n] = fmaxf(acc1[nt][i] + bias, 0.0f);
        }
    }
    __syncthreads();

    // ---- layer rho2: K=128 (32 steps), N=64 (4 tiles)
    v8f acc2[4];
#pragma unroll
    for (int nt = 0; nt < 4; ++nt)
        acc2[nt] = v8f{0.f, 0.f, 0.f, 0.f, 0.f, 0.f, 0.f, 0.f};

#pragma unroll 4
    for (int kk = 0; kk < 32; ++kk) {
        const int k0 = 4 * kk + 2 * half;
        v2f a = *(const v2f*)&sC[mloc * 128 + k0];
#pragma unroll
        for (int nt = 0; nt < 4; ++nt) {
            const int n = nt * 16 + n_lane;
            v2f b = v2f{Wr2[k0 * 64 + n], Wr2[(k0 + 1) * 64 + n]};
            acc2[nt] = wmma4(a, b, acc2[nt]);
        }
    }

#pragma unroll
    for (int nt = 0; nt < 4; ++nt) {
        const int n = nt * 16 + n_lane;
        const float bias = br2[n];
#pragma unroll
        for (int i = 0; i < 8; ++i) {
            const int r = rowbase + wave * 16 + 8 * half + i;
            agg[r * HDIM + n] = acc2[nt][i] + bias;   // last rho layer: no relu
        }
    }
}

// =====================================================================
// Head kernel: variant-selected 64->32(relu)->1 MLP, confidence scale,
// max_logit * tanh(logit / max_logit). One thread per row.
// =====================================================================
__global__ __launch_bounds__(256) void head_kernel(
    const float* __restrict__ agg,        // B x 64
    const int*  __restrict__ vtype,       // B
    const int*  __restrict__ altc,        // B
    const float* __restrict__ W0_0, const float* __restrict__ b0_0,
    const float* __restrict__ W1_0, const float* __restrict__ b1_0,
    const float* __restrict__ W0_1, const float* __restrict__ b0_1,
    const float* __restrict__ W1_1, const float* __restrict__ b1_1,
    const float* __restrict__ W0_2, const float* __restrict__ b0_2,
    const float* __restrict__ W1_2, const float* __restrict__ b1_2,
    const float* __restrict__ conf,       // 11
    const float* __restrict__ max_logit,  // 1
    float* __restrict__ out)              // B
{
    const int r = blockIdx.x * 256 + threadIdx.x;
    const int t = vtype[r];
    const float* W0 = (t == 0) ? W0_0 : (t == 1) ? W0_1 : W0_2;
    const float* b0 = (t == 0) ? b0_0 : (t == 1) ? b0_1 : b0_2;
    const float* W1 = (t == 0) ? W1_0 : (t == 1) ? W1_1 : W1_2;
    const float* b1 = (t == 0) ? b1_0 : (t == 1) ? b1_1 : b1_2;

    float x[HDIM];
#pragma unroll
    for (int k = 0; k < HDIM; ++k) x[k] = agg[r * HDIM + k];

    float logit = b1[0];
#pragma unroll 4
    for (int j = 0; j < 32; ++j) {
        float h = b0[j];
#pragma unroll
        for (int k = 0; k < HDIM; ++k) h = fmaf(x[k], W0[k * 32 + j], h);
        logit = fmaf(fmaxf(h, 0.0f), W1[j], logit);
    }

    const int a  = altc[r];
    const float c = conf[a < MAX_ALT ? a : MAX_ALT];
    const float ml = max_logit[0];
    const float L = logit * c;
    out[r] = ml * tanhf(L / ml);
}

// =====================================================================
// Launch. Input order assumed (setup_inputs insertion order, params leaves
// in insertion order): reads, info, seg_ids, variant_types, alt_counts,
// phi{W0,b0,W1,b1}, omega{W0,b0,W1,b1}, rho{W0,b0,W1,b1},
// outputs[0..2]{W0,b0,W1,b1}, confidence, max_logit  -> 31 tensors.
// Workspace: sums (NUM_SEG*64) | counts (NUM_SEG) | omega (B*64) | agg (B*64)
//            = ~16.2 MB of f32.
// =====================================================================
extern "C" void kernel_launch(void* const* d_in, const int* in_sizes, int n_in,
                              void* d_out, int out_size, void* d_ws, size_t ws_size,
                              hipStream_t stream) {
    const float* reads  = (const float*)d_in[0];
    const float* info   = (const float*)d_in[1];
    const int*   segid  = (const int*)d_in[2];
    const int*   vtype  = (const int*)d_in[3];
    const int*   altc   = (const int*)d_in[4];

    const float* phiW0  = (const float*)d_in[5];
    const float* phib0  = (const float*)d_in[6];
    const float* phiW1  = (const float*)d_in[7];
    const float* phib1  = (const float*)d_in[8];
    const float* omW0   = (const float*)d_in[9];
    const float* omb0   = (const float*)d_in[10];
    const float* omW1   = (const float*)d_in[11];
    const float* omb1   = (const float*)d_in[12];
    const float* rhoW0  = (const float*)d_in[13];
    const float* rhob0  = (const float*)d_in[14];
    const float* rhoW1  = (const float*)d_in[15];
    const float* rhob1  = (const float*)d_in[16];
    const float* o0W0 = (const float*)d_in[17]; const float* o0b0 = (const float*)d_in[18];
    const float* o0W1 = (const float*)d_in[19]; const float* o0b1 = (const float*)d_in[20];
    const float* o1W0 = (const float*)d_in[21]; const float* o1b0 = (const float*)d_in[22];
    const float* o1W1 = (const float*)d_in[23]; const float* o1b1 = (const float*)d_in[24];
    const float* o2W0 = (const float*)d_in[25]; const float* o2b0 = (const float*)d_in[26];
    const float* o2W1 = (const float*)d_in[27]; const float* o2b1 = (const float*)d_in[28];
    const float* conf   = (const float*)d_in[29];
    const float* mlogit = (const float*)d_in[30];

    float* ws     = (float*)d_ws;
    float* sums   = ws;                              // NUM_SEG * 64
    float* counts = sums + (size_t)NUM_SEG * HDIM;   // NUM_SEG
    float* omega  = counts + NUM_SEG;                // B * 64
    float* agg    = omega + (size_t)BATCH * HDIM;    // B * 64
    float* out    = (float*)d_out;

    // zero segment accumulators (graph-capture-safe async memset)
    hipMemsetAsync(d_ws, 0,
                   ((size_t)NUM_SEG * HDIM + NUM_SEG) * sizeof(float), stream);

    // phi encoder + segment scatter: 524288 reads / 128 per block
    encoder_kernel<NF_READ, true><<<TOTAL_READS / 128, 256, 0, stream>>>(
        reads, phiW0, phib0, phiW1, phib1, segid, sums, counts);

    // omega encoder: 16384 rows / 128 per block
    encoder_kernel<NF_INFO, false><<<BATCH / 128, 256, 0, stream>>>(
        info, omW0, omb0, omW1, omb1, nullptr, omega, nullptr);

    // rho: 16384 rows / 64 per block
    rho_kernel<<<BATCH / 64, 128, 0, stream>>>(
        sums, counts, omega, rhoW0, rhob0, rhoW1, rhob1, agg);

    // output heads + confidence + tanh clamp
    head_kernel<<<BATCH / 256, 256, 0, stream>>>(
        agg, vtype, altc,
        o0W0, o0b0, o0W1, o0b1,
        o1W0, o1b0, o1W1, o1b1,
        o2W0, o2b0, o2W1, o2b1,
        conf, mlogit, out);
}